// BackgroundNoiseLayer_4861902979700
// MI455X (gfx1250) — compile-verified
//
#include <hip/hip_runtime.h>
#include <stdint.h>

// ---- problem constants (from reference) ----
#define T_TOT 1000      // B*T timesteps
#define NBKG  100       // background units (K dimension)
#define NV1   50000
#define NLM   10000
#define NTOT  (NV1 + NLM)   // 60000 post neurons (N dimension)
#define CONN  4
#define REST_ELEMS (T_TOT * NBKG)   // 100000 floats

// ---- tiling ----
#define NPB      64          // neurons per block (4 wave-owned 16-wide N tiles)
#define THREADS  128         // 4 waves of 32
#define KPAIRS   (NBKG / 2)  // 50 pairs of K values
#define BP_STRIDE 80         // float2 stride per k-pair row in LDS B tile (padded: 64+16)
#define MTILES   ((T_TOT + 15) / 16)  // 63
#define ATILE    (16 * NBKG)          // 1600 floats per A tile

typedef __attribute__((ext_vector_type(2))) float v2f;
typedef __attribute__((ext_vector_type(8))) float v8f;

// ---- Tensor Data Mover availability / arity probing ----
#if defined(__has_builtin)
#  if __has_builtin(__builtin_amdgcn_tensor_load_to_lds)
#    define HAVE_TDM 1
#  endif
#endif
#if defined(__has_include)
#  if __has_include(<hip/amd_detail/amd_gfx1250_TDM.h>)
#    define TDM_SIX_ARGS 1   // therock headers => 6-arg builtin
#  endif
#endif

typedef unsigned int u32x4 __attribute__((ext_vector_type(4)));
typedef int          i32x4 __attribute__((ext_vector_type(4)));
typedef int          i32x8 __attribute__((ext_vector_type(8)));

#if defined(HAVE_TDM)
// Issue a 1-D TDM load of `tile_dim0`=ATILE f32 elements from gsrc into LDS at
// lds_off. elems_remaining = tensor_dim0: reads past it return zero (free
// zero-padding of the final partial M-tile).
__device__ __forceinline__ void tdm_load_tile(const float* gsrc, uint32_t lds_off,
                                              uint32_t elems_remaining) {
    const uint64_t ga = (uint64_t)(uintptr_t)gsrc;
    u32x4 g0;
    g0[0] = 1u;                                   // count=1, is_restore=0, no gather
    g0[1] = lds_off;                              // D#.lds_addr
    g0[2] = (uint32_t)(ga & 0xFFFFFFFFu);         // global_addr[31:0]
    g0[3] = (uint32_t)((ga >> 32) & 0x01FFFFFFu)  // global_addr[56:32]
            | (2u << 30);                         // type = 2 ("image")
    const uint32_t td0 = elems_remaining;
    i32x8 g1;
    g1[0] = 0x00020000;                           // data_size=2 (4B); no flags/pad
    g1[1] = (int)((td0 & 0xFFFFu) << 16);         // tensor_dim0[15:0]
    g1[2] = (int)((td0 >> 16) | (1u << 16));      // tensor_dim0[31:16]; tensor_dim1=1
    g1[3] = (int)((uint32_t)ATILE << 16);         // tile_dim0 = 1600
    g1[4] = 0;                                    // tile_dim1=0, tile_dim2=0 (1-D)
    g1[5] = ATILE;                                // tensor_dim0_stride
    g1[6] = 0;
    g1[7] = 0;
    i32x4 z4 = {0, 0, 0, 0};
#if defined(TDM_SIX_ARGS)
    i32x8 z8 = {0, 0, 0, 0, 0, 0, 0, 0};
    __builtin_amdgcn_tensor_load_to_lds(g0, g1, z4, z4, z8, 0);
#else
    __builtin_amdgcn_tensor_load_to_lds(g0, g1, z4, z4, 0);
#endif
}
#endif // HAVE_TDM

__global__ __launch_bounds__(THREADS)
void bkg_noise_wmma_kernel(const float* __restrict__ rest,
                           const float* __restrict__ v1_w,
                           const int*   __restrict__ v1_c,
                           const float* __restrict__ lm_w,
                           const int*   __restrict__ lm_c,
                           float* __restrict__ out)
{
    // B tile: Bt[pair p][n_local][j]; float2 row stride BP_STRIDE => lanes 0-15
    // and 16-31 hit disjoint bank halves on the b64 fragment loads.
    __shared__ float Bt[KPAIRS * BP_STRIDE * 2];   // 8000 floats = 32000 B
    // Double-buffered A tile: As[buf][m_local*NBKG + k]; stride 100 gives a
    // conflict-free bank pattern for the A fragment b64 loads.
    __shared__ float As[2][ATILE];                 // 2 * 6400 B

    const int tid   = threadIdx.x;
    const int nbase = blockIdx.x * NPB;

    // ---- zero the B tile ----
    for (int i = tid; i < KPAIRS * BP_STRIDE * 2; i += THREADS) Bt[i] = 0.0f;
    __syncthreads();

    // ---- scatter-add this block's COO entries into the LDS B tile ----
    // rows array is repeat(arange, CONN): neuron n owns entries [4n, 4n+4)
    for (int e = tid; e < NPB * CONN; e += THREADS) {
        const int nl = e >> 2;
        const int j  = e & 3;
        const int n  = nbase + nl;
        if (n < NTOT) {
            float w; int col;
            if (n < NV1) { const int idx = n * CONN + j;         w = v1_w[idx]; col = v1_c[idx]; }
            else         { const int idx = (n - NV1) * CONN + j; w = lm_w[idx]; col = lm_c[idx]; }
            atomicAdd(&Bt[((col >> 1) * BP_STRIDE + nl) * 2 + (col & 1)], w);  // ds_add_f32
        }
    }

    const int lane  = tid & 31;       // wave32
    const int wave  = tid >> 5;       // 0..3 : owns N sub-tile [nbase + wave*16, +16)
    const int lhalf = lane >> 4;      // 0 for lanes 0-15, 1 for lanes 16-31
    const int lmod  = lane & 15;

    const v2f* Bp = reinterpret_cast<const v2f*>(Bt);
    const int  n0 = nbase + wave * 16;
    const bool nvalid = (n0 + lmod) < NTOT;

#if defined(HAVE_TDM)
    // Prologue: wave 0 DMAs tile 0 into buffer 0 via the Tensor Data Mover.
    if (wave == 0) {
        tdm_load_tile(rest, (uint32_t)(uintptr_t)&As[0][0], REST_ELEMS);
        __builtin_amdgcn_s_wait_tensorcnt(0);
    }
#endif
    __syncthreads();

    for (int mt = 0; mt < MTILES; ++mt) {
        const int m0  = mt * 16;
        const int buf = mt & 1;

#if defined(HAVE_TDM)
        // Wave 0 kicks off the NEXT tile's DMA into the alternate buffer while
        // everyone computes the current one (overlap via TENSORcnt).
        if (wave == 0 && (mt + 1) < MTILES) {
            tdm_load_tile(rest + (mt + 1) * ATILE,
                          (uint32_t)(uintptr_t)&As[buf ^ 1][0],
                          (uint32_t)(REST_ELEMS - (mt + 1) * ATILE));
        }
#else
        // Fallback staging: the 16x100 tile is CONTIGUOUS in rest (row-major),
        // so copy it linearly with b128 vector loads, zero-padding the tail.
        {
            const int base  = m0 * NBKG;
            const int avail = REST_ELEMS - base;  // >= 800 for the last tile
            for (int i = tid * 4; i < ATILE; i += THREADS * 4) {
                float4 v;
                if (i + 3 < avail) {
                    v = *reinterpret_cast<const float4*>(rest + base + i);
                } else {
                    v.x = (i + 0 < avail) ? rest[base + i + 0] : 0.0f;
                    v.y = (i + 1 < avail) ? rest[base + i + 1] : 0.0f;
                    v.z = (i + 2 < avail) ? rest[base + i + 2] : 0.0f;
                    v.w = (i + 3 < avail) ? rest[base + i + 3] : 0.0f;
                }
                *reinterpret_cast<float4*>(&As[buf][i]) = v;
            }
        }
        __syncthreads();
#endif

        // ---- 25 x V_WMMA_F32_16X16X4_F32 over K = 100 ----
        const v2f* Ap = reinterpret_cast<const v2f*>(&As[buf][0]);
        v8f c = {};
        #pragma unroll
        for (int p0 = 0; p0 < KPAIRS; p0 += 2) {   // k0 = 2*p0
            // A frag: lanes 0-15 -> A[lmod][k0..k0+1], lanes 16-31 -> [k0+2..k0+3]
            v2f a = Ap[lmod * (NBKG / 2) + p0 + lhalf];
            // B frag: lanes 0-15 -> B[k0..k0+1][n], lanes 16-31 -> B[k0+2..k0+3][n]
            v2f b = Bp[(p0 + lhalf) * BP_STRIDE + wave * 16 + lmod];
            c = __builtin_amdgcn_wmma_f32_16x16x4_f32(
                    /*neg_a=*/false, a, /*neg_b=*/false, b,
                    /*c_mod=*/(short)0, c, /*reuse_a=*/false, /*reuse_b=*/false);
        }

        // ---- store 16x16 C tile (non-temporal: 240 MB stream, don't rinse L2) ----
        // VGPR r holds (M=r, lanes 0-15) and (M=r+8, lanes 16-31)
        #pragma unroll
        for (int r = 0; r < 8; ++r) {
            const int m = m0 + r + lhalf * 8;
            const int n = n0 + lmod;
            if (nvalid && m < T_TOT)
                __builtin_nontemporal_store(c[r], out + m * NTOT + n);
        }

#if defined(HAVE_TDM)
        if (wave == 0) __builtin_amdgcn_s_wait_tensorcnt(0);
#endif
        __syncthreads();   // next-tile buffer is ready / safe to overwrite
    }
}

extern "C" void kernel_launch(void* const* d_in, const int* in_sizes, int n_in,
                              void* d_out, int out_size, void* d_ws, size_t ws_size,
                              hipStream_t stream) {
    // setup_inputs() order: rest, v1_weights, v1_rows, v1_cols, lm_weights, lm_rows, lm_cols
    const float* rest = (const float*)d_in[0];
    const float* v1_w = (const float*)d_in[1];
    // d_in[2] = v1_rows: implicit (repeat(arange, CONN)), unused
    const int*   v1_c = (const int*)d_in[3];
    const float* lm_w = (const float*)d_in[4];
    // d_in[5] = lm_rows: implicit, unused
    const int*   lm_c = (const int*)d_in[6];
    float* out = (float*)d_out;

    const int nblocks = (NTOT + NPB - 1) / NPB;   // 938
    bkg_noise_wmma_kernel<<<nblocks, THREADS, 0, stream>>>(rest, v1_w, v1_c, lm_w, lm_c, out);
}